// fdnet_force_70970039599738
// MI455X (gfx1250) — compile-verified
//
#include <hip/hip_runtime.h>

typedef __attribute__((ext_vector_type(2))) float v2f;
typedef __attribute__((ext_vector_type(8))) float v8f;

#define XLEN 1024
#define BSZ  4096
#define FN   16
#define NONC 10
#define FD   8

// ws (float) layout:
//  [0..4]    c_int  : interior 5-tap coefficients (offsets -2..+2)
//  [8..12]   rowL0  : dev[0]   coeffs on x[0..4]
//  [16..20]  rowL1  : dev[1]   coeffs on x[0..4]
//  [24..28]  rowR2  : dev[L-2] coeffs on x[L-5..L-1]
//  [32..36]  rowR1  : dev[L-1] coeffs on x[L-5..L-1]
//  [64..64+XLEN)  bias[l] = wD0a[l,:] . w_fc1[2F:]

// B-matrix for the G composition GEMM:  Bm[q=(i*3+t)][n] = w1[i, n-t] (valid taps), n<5
__device__ __forceinline__ float bmval(int q, int n, const float* w1) {
  if (n >= 5) return 0.0f;
  int i = q / 3;
  int t = q - 3 * i;
  int s = n - t;
  return (s >= 0 && s <= 2) ? w1[i * 3 + s] : 0.0f;
}

// coeff of dif[f, pos] (pos=0,1,2) on x[j], j in 0..4 (left edge)
__device__ __forceinline__ float difL(const float* w1, const float* w1L, int f, int pos, int j) {
  if (pos == 0) return (j < 2) ? w1L[f * 2 + j] : 0.0f;
  if (pos == 1) return (j < 3) ? w1[f * 3 + j] : 0.0f;
  return (j >= 1 && j <= 3) ? w1[f * 3 + (j - 1)] : 0.0f;
}
// coeff of dif[f, L-3+p] (p=0,1,2) on x[L-5+j], j in 0..4 (right edge)
__device__ __forceinline__ float difR(const float* w1, const float* w1R, int f, int p, int j) {
  if (p == 0) return (j >= 1 && j <= 3) ? w1[f * 3 + (j - 1)] : 0.0f;
  if (p == 1) return (j >= 2) ? w1[f * 3 + (j - 2)] : 0.0f;
  return (j >= 3) ? w1R[f * 2 + (j - 3)] : 0.0f;
}

__global__ __launch_bounds__(256) void fdnet_precompute(
    const float* __restrict__ w1, const float* __restrict__ w1L,
    const float* __restrict__ w1R, const float* __restrict__ w2,
    const float* __restrict__ w2L, const float* __restrict__ w2R,
    const float* __restrict__ wfc, const float* __restrict__ wD0a,
    float* __restrict__ ws) {
  __shared__ float G[16][16];  // G[o][n]: 5-tap kernel of dif2 channel o (cols 0..4 used)
  const int tid = threadIdx.x;

  if (tid < 32) {
#if __has_builtin(__builtin_amdgcn_wmma_f32_16x16x4_f32)
    // G = A(16x48) * Bm(48x16), A[o][q] = w2 flat (row-major [o][i*3+t] == [o][q]).
    // 12 chained V_WMMA_F32_16X16X4_F32 over K.
    const int hh = tid >> 4;   // lane half selects K pair {0,1} vs {2,3}
    const int m  = tid & 15;   // A: row M; B: col N
    v8f acc = {0.f, 0.f, 0.f, 0.f, 0.f, 0.f, 0.f, 0.f};
#pragma unroll
    for (int kb = 0; kb < 12; ++kb) {
      const int q0 = 4 * kb + 2 * hh;
      v2f a, b;
      a.x = w2[m * 48 + q0];         // A[m][q0]
      a.y = w2[m * 48 + q0 + 1];     // A[m][q0+1]
      b.x = bmval(q0, m, w1);        // Bm[q0][m]
      b.y = bmval(q0 + 1, m, w1);    // Bm[q0+1][m]
      acc = __builtin_amdgcn_wmma_f32_16x16x4_f32(false, a, false, b, (short)0, acc,
                                                  false, false);
    }
#pragma unroll
    for (int v = 0; v < 8; ++v) G[v + 8 * hh][m] = acc[v];  // D layout: M=v(+8), N=lane%16
#else
    if (tid == 0) {
      for (int o = 0; o < 16; ++o)
        for (int n = 0; n < 16; ++n) {
          float s = 0.f;
          for (int q = 0; q < 48; ++q) s += w2[o * 48 + q] * bmval(q, n, w1);
          G[o][n] = s;
        }
    }
#endif
  }
  __syncthreads();

  if (tid == 0) {
    float af[16], bf[16];
    for (int f = 0; f < 16; ++f) { af[f] = wfc[2 * f]; bf[f] = wfc[2 * f + 1]; }
    // interior taps: c[n] = sum_o b_o*G[o][n] + sum_f a_f*w1[f][n-1] (n in 1..3)
    for (int n = 0; n < 5; ++n) {
      float c = 0.f;
      for (int o = 0; o < 16; ++o) c += bf[o] * G[o][n];
      if (n >= 1 && n <= 3)
        for (int f = 0; f < 16; ++f) c += af[f] * w1[f * 3 + (n - 1)];
      ws[n] = c;
    }
    // boundary rows
    for (int j = 0; j < 5; ++j) {
      float r0 = 0.f, r1 = 0.f, rr2 = 0.f, rr1 = 0.f;
      for (int f = 0; f < 16; ++f) {
        r0  += af[f] * difL(w1, w1L, f, 0, j);
        r1  += af[f] * difL(w1, w1L, f, 1, j);
        rr2 += af[f] * difR(w1, w1R, f, 1, j);
        rr1 += af[f] * difR(w1, w1R, f, 2, j);
      }
      for (int o = 0; o < 16; ++o) {
        float s0 = 0.f, s1 = 0.f, t2 = 0.f, t1 = 0.f;
        for (int i = 0; i < 16; ++i) {
          const float dl0 = difL(w1, w1L, i, 0, j);
          const float dl1 = difL(w1, w1L, i, 1, j);
          const float dl2 = difL(w1, w1L, i, 2, j);
          const float dr0 = difR(w1, w1R, i, 0, j);
          const float dr1 = difR(w1, w1R, i, 1, j);
          const float dr2 = difR(w1, w1R, i, 2, j);
          s0 += w2L[o * 32 + i * 2 + 0] * dl0 + w2L[o * 32 + i * 2 + 1] * dl1;
          s1 += w2[o * 48 + i * 3 + 0] * dl0 + w2[o * 48 + i * 3 + 1] * dl1 +
                w2[o * 48 + i * 3 + 2] * dl2;
          t2 += w2[o * 48 + i * 3 + 0] * dr0 + w2[o * 48 + i * 3 + 1] * dr1 +
                w2[o * 48 + i * 3 + 2] * dr2;
          t1 += w2R[o * 32 + i * 2 + 0] * dr1 + w2R[o * 32 + i * 2 + 1] * dr2;
        }
        r0 += bf[o] * s0; r1 += bf[o] * s1; rr2 += bf[o] * t2; rr1 += bf[o] * t1;
      }
      ws[8 + j] = r0; ws[16 + j] = r1; ws[24 + j] = rr2; ws[32 + j] = rr1;
    }
  }

  // bias[l] (batch-independent)
  for (int l = tid; l < XLEN; l += blockDim.x) {
    float s = 0.f;
    for (int n = 0; n < NONC; ++n) s += wD0a[l * NONC + n] * wfc[2 * FN + n];
    ws[64 + l] = s;
  }
}

__global__ __launch_bounds__(256) void fdnet_stencil(
    const float* __restrict__ xin, const float* __restrict__ ws,
    float* __restrict__ xout) {
  __shared__ float xs[8][XLEN + 4];  // one 4KB row per wave, +4 skew
  const int lane = threadIdx.x & 31;
  const int wv   = threadIdx.x >> 5;
  const long long row = (long long)blockIdx.x * 8 + wv;
  const float* src = xin + row * XLEN;
  float* rp = &xs[wv][0];

  // Async copy row -> LDS: 128 B per lane (8 x b128), fully coalesced across the wave.
  {
    unsigned lds0 = (unsigned)(size_t)(void*)rp + (unsigned)lane * 128u;
    const char* g0 = (const char*)src + (size_t)lane * 128;
#pragma unroll
    for (int i = 0; i < 8; ++i) {
      unsigned laddr = lds0 + 16u * i;
      const void* gaddr = (const void*)(g0 + 16 * i);
      asm volatile("global_load_async_to_lds_b128 %0, %1, off"
                   :: "v"(laddr), "v"(gaddr)
                   : "memory");
    }
  }

  // Coefficients (wave-uniform) and per-lane bias slice.
  const float c0 = ws[0], c1 = ws[1], c2 = ws[2], c3 = ws[3], c4 = ws[4];
  float lA[5], lB[5], rA[5], rB[5];
#pragma unroll
  for (int j = 0; j < 5; ++j) {
    lA[j] = ws[8 + j]; lB[j] = ws[16 + j]; rA[j] = ws[24 + j]; rB[j] = ws[32 + j];
  }
  float biasv[32];
#pragma unroll
  for (int k = 0; k < 32; ++k) biasv[k] = ws[64 + lane + 32 * k];

  asm volatile("s_wait_asynccnt 0x0" ::: "memory");
  __syncthreads();

  float nv[32];
  for (int it = 0; it < FD; ++it) {
#pragma unroll
    for (int k = 0; k < 32; ++k) {
      const int l = lane + 32 * k;
      float v;
      if (l >= 2 && l <= XLEN - 3) {
        v = c0 * rp[l - 2] + c1 * rp[l - 1] + c2 * rp[l] + c3 * rp[l + 1] + c4 * rp[l + 2];
      } else if (l == 0) {
        v = lA[0] * rp[0] + lA[1] * rp[1] + lA[2] * rp[2] + lA[3] * rp[3] + lA[4] * rp[4];
      } else if (l == 1) {
        v = lB[0] * rp[0] + lB[1] * rp[1] + lB[2] * rp[2] + lB[3] * rp[3] + lB[4] * rp[4];
      } else if (l == XLEN - 2) {
        v = rA[0] * rp[XLEN - 5] + rA[1] * rp[XLEN - 4] + rA[2] * rp[XLEN - 3] +
            rA[3] * rp[XLEN - 2] + rA[4] * rp[XLEN - 1];
      } else {  // l == XLEN-1
        v = rB[0] * rp[XLEN - 5] + rB[1] * rp[XLEN - 4] + rB[2] * rp[XLEN - 3] +
            rB[3] * rp[XLEN - 2] + rB[4] * rp[XLEN - 1];
      }
      nv[k] = rp[l] + v + biasv[k];
    }
    __syncthreads();  // all reads done before in-place update
#pragma unroll
    for (int k = 0; k < 32; ++k) rp[lane + 32 * k] = nv[k];
    __syncthreads();
  }

  // Final values are still in registers: coalesced stores straight to output.
  float* dst = xout + row * XLEN;
#pragma unroll
  for (int k = 0; k < 32; ++k) dst[lane + 32 * k] = nv[k];
}

extern "C" void kernel_launch(void* const* d_in, const int* in_sizes, int n_in,
                              void* d_out, int out_size, void* d_ws, size_t ws_size,
                              hipStream_t stream) {
  (void)in_sizes; (void)n_in; (void)out_size; (void)ws_size;
  const float* xIn  = (const float*)d_in[0];
  const float* w1   = (const float*)d_in[1];
  const float* w1L  = (const float*)d_in[2];
  const float* w1R  = (const float*)d_in[3];
  const float* w2   = (const float*)d_in[4];
  const float* w2L  = (const float*)d_in[5];
  const float* w2R  = (const float*)d_in[6];
  const float* wfc  = (const float*)d_in[7];
  const float* wD0a = (const float*)d_in[8];
  // d_in[9] = fdblock (== 8, compiled in as FD)
  float* ws  = (float*)d_ws;
  float* out = (float*)d_out;

  fdnet_precompute<<<1, 256, 0, stream>>>(w1, w1L, w1R, w2, w2L, w2R, wfc, wD0a, ws);
  fdnet_stencil<<<BSZ / 8, 256, 0, stream>>>(xIn, ws, out);
}